// GNN_47708496724689
// MI455X (gfx1250) — compile-verified
//
#include <hip/hip_runtime.h>

typedef __attribute__((ext_vector_type(2))) float v2f;
typedef __attribute__((ext_vector_type(8))) float v8f;

// ---------------- degree / norm kernels ----------------

__global__ void GNN_deg_init(float* deg_out, float* deg_in, int n) {
  int i = blockIdx.x * blockDim.x + threadIdx.x;
  if (i < n) { deg_out[i] = 1.0f; deg_in[i] = 1.0f; }  // self-loop contributes 1 to both
}

__global__ void GNN_deg_edges(const long long* __restrict__ src,
                              const long long* __restrict__ dst,
                              float* deg_out, float* deg_in, int e) {
  int i = blockIdx.x * blockDim.x + threadIdx.x;
  int stride = gridDim.x * blockDim.x;
  for (; i < e; i += stride) {
    atomicAdd(&deg_out[(int)src[i]], 1.0f);
    atomicAdd(&deg_in[(int)dst[i]], 1.0f);
  }
}

__global__ void GNN_deg_to_norm(float* a, float* b, int n) {
  int i = blockIdx.x * blockDim.x + threadIdx.x;
  if (i < n) {
    a[i] = rsqrtf(fmaxf(a[i], 1.0f));
    b[i] = rsqrtf(fmaxf(b[i], 1.0f));
  }
}

// ---------------- WMMA GEMMs (f32 16x16x4) ----------------
// A-frag (16x4 f32): lanes 0-15 rows M=0..15; VGPR0/1 = K {0,1} (lanes 0-15) or {2,3} (lanes 16-31)
// B-frag (4x16 f32): lanes index N; VGPR0/1 = K {0,1} (lanes 0-15) or {2,3} (lanes 16-31)
// C/D (16x16 f32):   VGPR v, lanes 0-15 -> (M=v, N=lane); lanes 16-31 -> (M=v+8, N=lane-16)

// xw = (x * norm_src[:,None]) @ W1 ; dual-store into xw1 and agg1 (self-loop init)
__global__ void GNN_gemm1(const float* __restrict__ x, const float* __restrict__ ns,
                          const float* __restrict__ W, float* __restrict__ xw,
                          float* __restrict__ agg, int N) {
  const int wave = threadIdx.x >> 5;       // 8 waves -> 8 col tiles of 16 (out width 128)
  const int lane = threadIdx.x & 31;
  const int half = lane >> 4;
  const int l    = lane & 15;
  const int row0 = blockIdx.x * 16;
  const int col0 = wave * 16;

  int arow_idx = row0 + l;
  if (arow_idx >= N) arow_idx = N - 1;     // clamp loads; stores guarded below
  const float* arow = x + (size_t)arow_idx * 128;
  const float scale = ns[arow_idx];

  v8f c = {};
  #pragma unroll
  for (int k = 0; k < 128; k += 4) {
    const int ka = k + 2 * half;
    v2f a, b;
    a.x = arow[ka] * scale;
    a.y = arow[ka + 1] * scale;
    b.x = W[(size_t)ka * 128 + col0 + l];
    b.y = W[(size_t)(ka + 1) * 128 + col0 + l];
    c = __builtin_amdgcn_wmma_f32_16x16x4_f32(false, a, false, b, (short)0, c,
                                              false, false);
  }
  #pragma unroll
  for (int v = 0; v < 8; ++v) {
    const int row = row0 + v + 8 * half;
    if (row < N) {
      const size_t idx = (size_t)row * 128 + col0 + l;
      xw[idx]  = c[v];
      agg[idx] = c[v];
    }
  }
}

// xw2 = h2 @ W2 (128 -> 64) ; dual-store into xw2 and agg2
__global__ void GNN_gemm2(const float* __restrict__ h, const float* __restrict__ W,
                          float* __restrict__ xw, float* __restrict__ agg, int N) {
  const int wave = threadIdx.x >> 5;       // 8 waves: 2 row tiles x 4 col tiles
  const int lane = threadIdx.x & 31;
  const int half = lane >> 4;
  const int l    = lane & 15;
  const int row0 = blockIdx.x * 32 + (wave >> 2) * 16;
  const int col0 = (wave & 3) * 16;

  int arow_idx = row0 + l;
  if (arow_idx >= N) arow_idx = N - 1;
  const float* arow = h + (size_t)arow_idx * 128;

  v8f c = {};
  #pragma unroll
  for (int k = 0; k < 128; k += 4) {
    const int ka = k + 2 * half;
    v2f a, b;
    a.x = arow[ka];
    a.y = arow[ka + 1];
    b.x = W[(size_t)ka * 64 + col0 + l];
    b.y = W[(size_t)(ka + 1) * 64 + col0 + l];
    c = __builtin_amdgcn_wmma_f32_16x16x4_f32(false, a, false, b, (short)0, c,
                                              false, false);
  }
  #pragma unroll
  for (int v = 0; v < 8; ++v) {
    const int row = row0 + v + 8 * half;
    if (row < N) {
      const size_t idx = (size_t)row * 64 + col0 + l;
      xw[idx]  = c[v];
      agg[idx] = c[v];
    }
  }
}

// ---------------- edge aggregation (one wave per edge) ----------------

__global__ void GNN_agg128(const long long* __restrict__ src,
                           const long long* __restrict__ dst,
                           const float* __restrict__ xw, float* __restrict__ agg,
                           int E) {
  const int gwave  = (int)((blockIdx.x * blockDim.x + threadIdx.x) >> 5);
  const int lane   = threadIdx.x & 31;
  const int nwaves = (int)((gridDim.x * blockDim.x) >> 5);
  for (int e = gwave; e < E; e += nwaves) {
    const int s = (int)src[e];
    const int d = (int)dst[e];
    const float4 v = ((const float4*)(xw + (size_t)s * 128))[lane];  // coalesced 512B/edge
    float* ap = agg + (size_t)d * 128 + lane * 4;
    atomicAdd(ap + 0, v.x);
    atomicAdd(ap + 1, v.y);
    atomicAdd(ap + 2, v.z);
    atomicAdd(ap + 3, v.w);
  }
}

__global__ void GNN_agg64(const long long* __restrict__ src,
                          const long long* __restrict__ dst,
                          const float* __restrict__ xw, float* __restrict__ agg,
                          int E) {
  const int gwave  = (int)((blockIdx.x * blockDim.x + threadIdx.x) >> 5);
  const int lane   = threadIdx.x & 31;
  const int nwaves = (int)((gridDim.x * blockDim.x) >> 5);
  for (int e = gwave; e < E; e += nwaves) {
    const int s = (int)src[e];
    const int d = (int)dst[e];
    const float2 v = ((const float2*)(xw + (size_t)s * 64))[lane];
    float* ap = agg + (size_t)d * 64 + lane * 2;
    atomicAdd(ap + 0, v.x);
    atomicAdd(ap + 1, v.y);
  }
}

// ---------------- elementwise fusions ----------------

// h2 = relu(agg1 * norm_dst + b1) * norm_src   (pre-scale for layer-2 GEMM), in place
__global__ void GNN_relu_scale(float* __restrict__ buf, const float* __restrict__ nd,
                               const float* __restrict__ ns, const float* __restrict__ b1,
                               int total) {
  int i = blockIdx.x * blockDim.x + threadIdx.x;
  if (i < total) {
    const int node = i >> 7;       // /128
    const int f    = i & 127;
    float v = buf[i] * nd[node] + b1[f];
    v = fmaxf(v, 0.0f);
    buf[i] = v * ns[node];
  }
}

// out = agg2 * norm_dst + b2
__global__ void GNN_final(const float* __restrict__ agg, const float* __restrict__ nd,
                          const float* __restrict__ b2, float* __restrict__ out,
                          int total) {
  int i = blockIdx.x * blockDim.x + threadIdx.x;
  if (i < total) {
    const int node = i >> 6;       // /64
    const int f    = i & 63;
    out[i] = agg[i] * nd[node] + b2[f];
  }
}

// ---------------- host-side orchestration ----------------

extern "C" void kernel_launch(void* const* d_in, const int* in_sizes, int n_in,
                              void* d_out, int out_size, void* d_ws, size_t ws_size,
                              hipStream_t stream) {
  const float*     x   = (const float*)d_in[0];
  const long long* src = (const long long*)d_in[1];
  const long long* dst = (const long long*)d_in[2];
  const float*     W1  = (const float*)d_in[3];
  const float*     b1  = (const float*)d_in[4];
  const float*     W2  = (const float*)d_in[5];
  const float*     b2  = (const float*)d_in[6];

  const int N = in_sizes[0] / 128;
  const int E = in_sizes[1];

  float* ws       = (float*)d_ws;
  float* norm_src = ws;                       // N floats (deg_out -> norm)
  float* norm_dst = ws + N;                   // N floats (deg_in  -> norm)
  float* buf1     = norm_dst + N;             // N*128: xw1, later xw2(+agg2)
  float* buf2     = buf1 + (size_t)N * 128;   // N*128: agg1 -> h2 (in place)

  float* xw1  = buf1;
  float* agg1 = buf2;
  float* h2   = buf2;
  float* xw2  = buf1;
  float* agg2 = buf1 + (size_t)N * 64;

  float* out = (float*)d_out;

  const int TB = 256;
  const int nBlocksN    = (N + TB - 1) / TB;
  const int nBlocksE    = (E + TB - 1) / TB;
  const int edgeWaveBlk = (E + 7) / 8;        // one wave32 per edge, 8 waves/block
  const int gemm1Blk    = (N + 15) / 16;
  const int gemm2Blk    = (N + 31) / 32;

  // degrees & norms
  GNN_deg_init<<<nBlocksN, TB, 0, stream>>>(norm_src, norm_dst, N);
  GNN_deg_edges<<<nBlocksE, TB, 0, stream>>>(src, dst, norm_src, norm_dst, E);
  GNN_deg_to_norm<<<nBlocksN, TB, 0, stream>>>(norm_src, norm_dst, N);

  // layer 1
  GNN_gemm1<<<gemm1Blk, TB, 0, stream>>>(x, norm_src, W1, xw1, agg1, N);
  GNN_agg128<<<edgeWaveBlk, TB, 0, stream>>>(src, dst, xw1, agg1, E);
  GNN_relu_scale<<<((size_t)N * 128 + TB - 1) / TB, TB, 0, stream>>>(
      agg1, norm_dst, norm_src, b1, N * 128);

  // layer 2
  GNN_gemm2<<<gemm2Blk, TB, 0, stream>>>(h2, W2, xw2, agg2, N);
  GNN_agg64<<<edgeWaveBlk, TB, 0, stream>>>(src, dst, xw2, agg2, E);
  GNN_final<<<((size_t)N * 64 + TB - 1) / TB, TB, 0, stream>>>(
      agg2, norm_dst, b2, out, N * 64);
}